// Model_3985729651446
// MI455X (gfx1250) — compile-verified
//
#include <hip/hip_runtime.h>

// ---------------------------------------------------------------------------
// GraphSAGE (3 layers) + edge predictor for MI455X (gfx1250, wave32, WMMA).
// All dense math through V_WMMA_F32_16X16X4_F32 (fp32; workload is
// gather/atomic bound, so fp32 matrix path keeps reference precision for free).
//
// Fragment layouts (CDNA5 ISA 7.12.2):
//   A 16x4 : lane-half h, M=lane%16, VGPR0=K(2h), VGPR1=K(2h+1)
//   B 4x16 : lane-half h, N=lane%16, VGPR0=K(2h), VGPR1=K(2h+1)  -> one b64 load
//   C 16x16: VGPR r, lanes0-15 M=r, lanes16-31 M=r+8, N=lane%16
// Invalid (padded) columns load clamped weight rows unconditionally and are
// simply never stored -> EXEC stays all-ones (WMMA requirement), no exec-mask
// save/restore churn.
// ---------------------------------------------------------------------------

typedef float v2f __attribute__((ext_vector_type(2)));
typedef float v8f __attribute__((ext_vector_type(8)));

__device__ __forceinline__ v8f wmma_f32(v2f a, v2f b, v8f c) {
  return __builtin_amdgcn_wmma_f32_16x16x4_f32(false, a, false, b,
                                               (short)0, c, false, false);
}

// -------------------------------- utilities --------------------------------

__global__ void zero_kernel(float* __restrict__ p, int n) {
  int i = blockIdx.x * blockDim.x + threadIdx.x;
  if (i < n) p[i] = 0.0f;
}

__global__ void deg_kernel(const int* __restrict__ dst, float* __restrict__ deg, int E_) {
  int i = blockIdx.x * blockDim.x + threadIdx.x;
  if (i < E_) atomicAdd(&deg[dst[i]], 1.0f);
}

__global__ void invdeg_kernel(const float* __restrict__ deg, float* __restrict__ invdeg, int n) {
  int i = blockIdx.x * blockDim.x + threadIdx.x;
  if (i < n) {
    float d = deg[i];
    invdeg[i] = (d > 0.0f) ? (1.0f / d) : 0.0f;   // 0 marks isolated node
  }
}

// ------------------------- per-edge message GEMM ---------------------------
// agg[dst[e]] += concat(h[src[e]], efeat[e]) @ Wm^T   (bias folded out).
// One wave: 16 edges x 32 out-features (2 accumulators), K-loop fully
// unrolled in steps of 4.

template <int DH, int DE, int DOUT>
__global__ __launch_bounds__(256)
void msg_kernel(const float* __restrict__ h,       // [N, DH]
                const float* __restrict__ efeat,   // [E, DE]
                const int*   __restrict__ src,
                const int*   __restrict__ dst,
                const float* __restrict__ Wm,      // [DOUT, DH+DE] row-major
                float*       __restrict__ agg,     // [N, DOUT]
                int E_) {
  constexpr int DIN   = DH + DE;
  static_assert(DH % 4 == 0 && DIN % 4 == 0, "K tiling needs /4 dims");
  constexpr int NT    = (DOUT + 15) / 16;
  constexpr int NPAIR = (NT + 1) / 2;

  const int wave  = blockIdx.x * (blockDim.x >> 5) + (threadIdx.x >> 5);
  const int pair  = wave % NPAIR;
  const int etile = wave / NPAIR;
  if (etile * 16 >= E_) return;

  const int lane = threadIdx.x & 31;
  const int half = lane >> 4;
  const int m    = lane & 15;

  const int e = min(etile * 16 + m, E_ - 1);
  const int s = src[e];
  const float* hrow = h + (long)s * DH;
  const float* erow = efeat + (long)e * DE;

  const int  n0  = pair * 32 + m;
  const int  n1  = n0 + 16;
  const bool nv0 = (n0 < DOUT);
  const bool nv1 = (n1 < DOUT);
  const float* wrow0 = Wm + (long)(nv0 ? n0 : 0) * DIN;
  const float* wrow1 = Wm + (long)(nv1 ? n1 : 0) * DIN;

  v8f c0 = {}, c1 = {};
#pragma unroll
  for (int k0 = 0; k0 < DIN; k0 += 4) {
    const int kb = k0 + 2 * half;                      // compile-time offset
    const float* ap = (kb < DH) ? (hrow + kb) : (erow + (kb - DH));
    v2f a  = *(const v2f*)ap;
    v2f b0 = *(const v2f*)(wrow0 + kb);
    v2f b1 = *(const v2f*)(wrow1 + kb);
    c0 = wmma_f32(a, b0, c0);
    c1 = wmma_f32(a, b1, c1);
  }

#pragma unroll
  for (int r = 0; r < 8; ++r) {                        // C row: M = r + 8*half
    const int em = etile * 16 + r + 8 * half;
    if (em < E_) {
      const long drow = (long)dst[em] * DOUT;
      if (nv0) atomicAdd(&agg[drow + n0], c0[r]);
      if (nv1) atomicAdd(&agg[drow + n1], c1[r]);
    }
  }
}

// ---------------------------- node apply GEMM ------------------------------
// hout = relu( concat(h, agg*invdeg + bm) @ Wa^T + ba ); isolated nodes get 0
// neighbor half (bsel masks the bias term).

template <int DH, int DNB, int DOUT>
__global__ __launch_bounds__(256)
void apply_kernel(const float* __restrict__ h,        // [N, DH]
                  const float* __restrict__ agg,      // [N, DNB]
                  const float* __restrict__ invdeg,   // [N]
                  const float* __restrict__ bm,       // [DNB]
                  const float* __restrict__ Wa,       // [DOUT, DH+DNB]
                  const float* __restrict__ ba,       // [DOUT]
                  float*       __restrict__ hout,     // [N, DOUT]
                  int N_) {
  constexpr int DIN   = DH + DNB;
  static_assert(DH % 4 == 0 && DIN % 4 == 0, "K tiling needs /4 dims");
  constexpr int NT    = (DOUT + 15) / 16;
  constexpr int NPAIR = (NT + 1) / 2;

  const int wave  = blockIdx.x * (blockDim.x >> 5) + (threadIdx.x >> 5);
  const int pair  = wave % NPAIR;
  const int mtile = wave / NPAIR;
  if (mtile * 16 >= N_) return;

  const int lane = threadIdx.x & 31;
  const int half = lane >> 4;
  const int m    = lane & 15;

  const int node = min(mtile * 16 + m, N_ - 1);
  const float iv   = invdeg[node];
  const float bsel = (iv > 0.0f) ? 1.0f : 0.0f;
  const float* hrow = h + (long)node * DH;
  const float* arow = agg + (long)node * DNB;

  const int  n0  = pair * 32 + m;
  const int  n1  = n0 + 16;
  const bool nv0 = (n0 < DOUT);
  const bool nv1 = (n1 < DOUT);
  const float* wrow0 = Wa + (long)(nv0 ? n0 : 0) * DIN;
  const float* wrow1 = Wa + (long)(nv1 ? n1 : 0) * DIN;

  v8f c0 = {}, c1 = {};
#pragma unroll
  for (int k0 = 0; k0 < DIN; k0 += 4) {
    const int kb = k0 + 2 * half;
    v2f a;
    if (kb < DH) {                                     // compile-time branch
      a = *(const v2f*)(hrow + kb);
    } else {
      v2f g  = *(const v2f*)(arow + (kb - DH));
      v2f bb = *(const v2f*)(bm + (kb - DH));
      a.x = g.x * iv + bb.x * bsel;                    // 0 for isolated nodes
      a.y = g.y * iv + bb.y * bsel;
    }
    v2f b0 = *(const v2f*)(wrow0 + kb);
    v2f b1 = *(const v2f*)(wrow1 + kb);
    c0 = wmma_f32(a, b0, c0);
    c1 = wmma_f32(a, b1, c1);
  }

  const float bias0 = nv0 ? ba[n0] : 0.0f;
  const float bias1 = nv1 ? ba[n1] : 0.0f;
#pragma unroll
  for (int r = 0; r < 8; ++r) {
    const int nr = mtile * 16 + r + 8 * half;
    if (nr < N_) {
      if (nv0) {
        float v = c0[r] + bias0;
        hout[(long)nr * DOUT + n0] = v > 0.0f ? v : 0.0f;
      }
      if (nv1) {
        float v = c1[r] + bias1;
        hout[(long)nr * DOUT + n1] = v > 0.0f ? v : 0.0f;
      }
    }
  }
}

// ----------------------------- edge predictor ------------------------------
// out[e] = concat(h3[src[e]], h3[dst[e]]) @ Wp^T + bp   (15 classes, 1 ntile)

template <int DF, int NC>
__global__ __launch_bounds__(256)
void pred_kernel(const float* __restrict__ h,    // [N, DF]
                 const int*   __restrict__ src,
                 const int*   __restrict__ dst,
                 const float* __restrict__ Wp,   // [NC, 2*DF]
                 const float* __restrict__ bp,   // [NC]
                 float*       __restrict__ out,  // [E, NC]
                 int E_) {
  constexpr int DIN = 2 * DF;
  const int etile = blockIdx.x * (blockDim.x >> 5) + (threadIdx.x >> 5);
  if (etile * 16 >= E_) return;

  const int lane = threadIdx.x & 31;
  const int half = lane >> 4;
  const int m    = lane & 15;

  const int e = min(etile * 16 + m, E_ - 1);
  const float* srow = h + (long)src[e] * DF;
  const float* drow = h + (long)dst[e] * DF;

  const int  n  = m;
  const bool nv = (n < NC);
  const float* wrow = Wp + (long)(nv ? n : 0) * DIN;

  v8f c = {};
#pragma unroll
  for (int k0 = 0; k0 < DIN; k0 += 4) {
    const int kb = k0 + 2 * half;
    const float* ap = (kb < DF) ? (srow + kb) : (drow + (kb - DF));
    v2f a = *(const v2f*)ap;
    v2f b = *(const v2f*)(wrow + kb);
    c = wmma_f32(a, b, c);
  }

  if (nv) {
    const float bias = bp[n];
#pragma unroll
    for (int r = 0; r < 8; ++r) {
      const int em = etile * 16 + r + 8 * half;
      if (em < E_) out[(long)em * NC + n] = c[r] + bias;
    }
  }
}

// ------------------------------- launcher ----------------------------------

extern "C" void kernel_launch(void* const* d_in, const int* in_sizes, int n_in,
                              void* d_out, int out_size, void* d_ws, size_t ws_size,
                              hipStream_t stream) {
  constexpr int NDIM = 128;   // ndim_in / edim / ndim_out
  constexpr int HID  = 152;
  constexpr int NCLS = 15;

  const float* nfeats = (const float*)d_in[0];   // [N,1,128]
  const float* efeats = (const float*)d_in[1];   // [E,1,128]
  const int*   src    = (const int*)d_in[2];
  const int*   dst    = (const int*)d_in[3];
  const float* Wm1 = (const float*)d_in[4];  const float* bm1 = (const float*)d_in[5];
  const float* Wa1 = (const float*)d_in[6];  const float* ba1 = (const float*)d_in[7];
  const float* Wm2 = (const float*)d_in[8];  const float* bm2 = (const float*)d_in[9];
  const float* Wa2 = (const float*)d_in[10]; const float* ba2 = (const float*)d_in[11];
  const float* Wm3 = (const float*)d_in[12]; const float* bm3 = (const float*)d_in[13];
  const float* Wa3 = (const float*)d_in[14]; const float* ba3 = (const float*)d_in[15];
  const float* Wp  = (const float*)d_in[16]; const float* bp  = (const float*)d_in[17];

  const int N = in_sizes[0] / NDIM;
  const int E = in_sizes[2];

  // workspace carve-up (floats)
  float* w      = (float*)d_ws;
  float* hA     = w;                       // [N,152]
  float* hB     = hA  + (size_t)N * HID;   // [N,152]
  float* h3     = hB  + (size_t)N * HID;   // [N,128]
  float* agg    = h3  + (size_t)N * NDIM;  // [N,152]
  float* deg    = agg + (size_t)N * HID;   // [N]
  float* invdeg = deg + N;                 // [N]

  auto cdiv = [](int a, int b) { return (a + b - 1) / b; };
  const int WPB = 8;   // 256-thread blocks = 8 waves

  // degrees (shared across layers)
  zero_kernel<<<cdiv(N, 256), 256, 0, stream>>>(deg, N);
  deg_kernel<<<cdiv(E, 256), 256, 0, stream>>>(dst, deg, E);
  invdeg_kernel<<<cdiv(N, 256), 256, 0, stream>>>(deg, invdeg, N);

  // ---- layer 1: msg(128+128 -> 152), apply(128+152 -> 152) ----
  zero_kernel<<<cdiv(N * HID, 256), 256, 0, stream>>>(agg, N * HID);
  {
    constexpr int NP = ((HID + 15) / 16 + 1) / 2;     // 5 pairs of n-tiles
    int tiles = cdiv(E, 16) * NP;
    msg_kernel<NDIM, NDIM, HID><<<cdiv(tiles, WPB), 256, 0, stream>>>(
        nfeats, efeats, src, dst, Wm1, agg, E);
    int atiles = cdiv(N, 16) * NP;
    apply_kernel<NDIM, HID, HID><<<cdiv(atiles, WPB), 256, 0, stream>>>(
        nfeats, agg, invdeg, bm1, Wa1, ba1, hA, N);
  }

  // ---- layer 2: msg(152+128 -> 152), apply(152+152 -> 152) ----
  zero_kernel<<<cdiv(N * HID, 256), 256, 0, stream>>>(agg, N * HID);
  {
    constexpr int NP = ((HID + 15) / 16 + 1) / 2;
    int tiles = cdiv(E, 16) * NP;
    msg_kernel<HID, NDIM, HID><<<cdiv(tiles, WPB), 256, 0, stream>>>(
        hA, efeats, src, dst, Wm2, agg, E);
    int atiles = cdiv(N, 16) * NP;
    apply_kernel<HID, HID, HID><<<cdiv(atiles, WPB), 256, 0, stream>>>(
        hA, agg, invdeg, bm2, Wa2, ba2, hB, N);
  }

  // ---- layer 3: msg(152+128 -> 128), apply(152+128 -> 128) ----
  zero_kernel<<<cdiv(N * NDIM, 256), 256, 0, stream>>>(agg, N * NDIM);
  {
    constexpr int NP = ((NDIM + 15) / 16 + 1) / 2;    // 4 pairs
    int tiles = cdiv(E, 16) * NP;
    msg_kernel<HID, NDIM, NDIM><<<cdiv(tiles, WPB), 256, 0, stream>>>(
        hB, efeats, src, dst, Wm3, agg, E);
    int atiles = cdiv(N, 16) * NP;
    apply_kernel<HID, NDIM, NDIM><<<cdiv(atiles, WPB), 256, 0, stream>>>(
        hB, agg, invdeg, bm3, Wa3, ba3, h3, N);
  }

  // ---- edge predictor ----
  {
    int tiles = cdiv(E, 16);   // single 16-wide class tile (15 valid)
    pred_kernel<NDIM, NCLS><<<cdiv(tiles, WPB), 256, 0, stream>>>(
        h3, src, dst, Wp, bp, (float*)d_out, E);
  }
}